// hgnn_38800734552596
// MI455X (gfx1250) — compile-verified
//
#include <hip/hip_runtime.h>
#include <hip/hip_bf16.h>

#define N_NODES  5200
#define N_INCID  5200
#define N_HEDGES 1000
#define N_DIM    4096
#define NEG_SLOPE 0.01f

typedef float v2f __attribute__((ext_vector_type(2)));
typedef float v8f __attribute__((ext_vector_type(8)));

// ---------------------------------------------------------------- init
__global__ void hg_init(float* D, float* B, int* ecnt, int* ncnt) {
    int i = blockIdx.x * blockDim.x + threadIdx.x;
    if (i < N_NODES)  { D[i] = 0.f; ncnt[i] = 0; }
    if (i < N_HEDGES) { B[i] = 0.f; ecnt[i] = 0; }
}

// ---------------------------------------------------------------- degree counts
__global__ void hg_count(const int* node_idx, const int* edge_idx,
                         const float* hw, const float* norm,
                         float* D, float* B, int* ecnt, int* ncnt) {
    int i = blockIdx.x * blockDim.x + threadIdx.x;
    if (i >= N_INCID) return;
    int n = node_idx[i];
    int e = edge_idx[i];
    atomicAdd(&D[n], hw[e]);        // D = segment_sum(hw[edge_idx], node_idx)
    atomicAdd(&B[e], norm[n]);      // B = segment_sum(norm[node_idx], edge_idx)
    atomicAdd(&ecnt[e], 1);
    atomicAdd(&ncnt[n], 1);
}

// ---------------------------------------------------------------- single-block exclusive scan
#define SCAN_MAX 5216
__global__ void hg_scan(const int* cnt, int* off, int* cur, int n) {
    __shared__ int sa[SCAN_MAX];
    __shared__ int sb[SCAN_MAX];
    int tid = threadIdx.x;
    for (int i = tid; i < n; i += blockDim.x) sa[i] = cnt[i];
    __syncthreads();
    int* src = sa; int* dst = sb;
    for (int ofs = 1; ofs < n; ofs <<= 1) {
        for (int i = tid; i < n; i += blockDim.x)
            dst[i] = src[i] + ((i >= ofs) ? src[i - ofs] : 0);
        __syncthreads();
        int* t = src; src = dst; dst = t;
    }
    for (int i = tid; i < n; i += blockDim.x) {
        int o = (i == 0) ? 0 : src[i - 1];
        off[i] = o;
        cur[i] = o;
    }
    if (tid == 0) off[n] = src[n - 1];
}

// ---------------------------------------------------------------- CSR placement (order fixed later)
__global__ void hg_place(const int* node_idx, const int* edge_idx,
                         int* ecur, int* ncur, int* eperm, int* nperm) {
    int i = blockIdx.x * blockDim.x + threadIdx.x;
    if (i >= N_INCID) return;
    int e = edge_idx[i];
    int pe = atomicAdd(&ecur[e], 1);
    eperm[pe] = i;
    int n = node_idx[i];
    int pn = atomicAdd(&ncur[n], 1);
    nperm[pn] = i;
}

// ---------------------------------------------------------------- per-segment sort -> deterministic order
__global__ void hg_sortseg(int* eperm, const int* eoff, int* nperm, const int* noff) {
    int t = blockIdx.x * blockDim.x + threadIdx.x;
    int* perm; const int* off; int s;
    if (t < N_HEDGES)                 { perm = eperm; off = eoff; s = t; }
    else if (t < N_HEDGES + N_NODES)  { perm = nperm; off = noff; s = t - N_HEDGES; }
    else return;
    int b0 = off[s], b1 = off[s + 1];
    for (int i = b0 + 1; i < b1; ++i) {
        int v = perm[i];
        int j = i - 1;
        while (j >= b0 && perm[j] > v) { perm[j + 1] = perm[j]; --j; }
        perm[j + 1] = v;
    }
}

// ---------------------------------------------------------------- edge phase: WMMA f32 16x16x4 segment-sum
// One wave owns a 16-hedge x 16-dim fp32 C tile. CSR members of the 16 hedges
// (contiguous in eperm) are consumed 4 at a time: A(16x4) is the one-hot
// scatter matrix carrying Binv[h], B(4x16) the gathered feature fragment.
// Main loop is clamp-free (no EXEC divergence around v_wmma); only the last
// (partial) chunk takes the masked tail path, using bitwise & (no branches).
__global__ void hg_edge_wmma(const float* __restrict__ feat,
                             const int* __restrict__ node_idx,
                             const int* __restrict__ edge_idx,
                             const int* __restrict__ eoff,
                             const int* __restrict__ eperm,
                             const float* __restrict__ Bsum,
                             float* __restrict__ edge_raw) {
    int group = blockIdx.x;                 // 16 hedges per group
    int h0 = group * 16;
    int lane = threadIdx.x & 31;
    int wave = threadIdx.x >> 5;            // 4 waves / block
    int dimbase = (blockIdx.y * 4 + wave) * 16;

    int hend = (h0 + 16 < N_HEDGES) ? (h0 + 16) : N_HEDGES;
    int base = eoff[h0];
    int pend = eoff[hend];
    int nmemb = pend - base;
    int pfull = base + (nmemb & ~3);        // last full-chunk boundary

    int m = lane & 15;                      // A row (hedge slot) == B col (dim)
    int h = h0 + m;
    float binv = 0.f;
    if (h < N_HEDGES) {
        float b = Bsum[h];
        binv = (b == 0.f) ? 0.f : (1.f / b);
    }
    int klo = (lane < 16) ? 0 : 2;          // this half-wave supplies K = klo, klo+1
    const float* fcol = feat + dimbase + m; // column slice of features

    v8f c = {};                             // 16x16 accumulator, M=r / r+8 per ISA layout
    int p = base;
    #pragma unroll 2
    for (; p < pfull; p += 4) {
        int j0 = eperm[p + klo];
        int j1 = eperm[p + klo + 1];
        int s0 = edge_idx[j0];
        int s1 = edge_idx[j1];
        int n0 = node_idx[j0];
        int n1 = node_idx[j1];
        v2f a, b;
        a.x = (s0 == h) ? binv : 0.f;
        a.y = (s1 == h) ? binv : 0.f;
        b.x = fcol[(size_t)n0 * N_DIM];
        b.y = fcol[(size_t)n1 * N_DIM];
        c = __builtin_amdgcn_wmma_f32_16x16x4_f32(false, a, false, b,
                                                  (short)0, c, false, false);
    }
    if (p < pend) {                         // single masked tail chunk
        int p0 = p + klo;
        int p1 = p0 + 1;
        int q0 = (p0 < pend) ? p0 : (pend - 1);
        int q1 = (p1 < pend) ? p1 : (pend - 1);
        int j0 = eperm[q0];
        int j1 = eperm[q1];
        int s0 = edge_idx[j0];
        int s1 = edge_idx[j1];
        int n0 = node_idx[j0];
        int n1 = node_idx[j1];
        v2f a, b;
        a.x = (((int)(p0 < pend)) & ((int)(s0 == h))) ? binv : 0.f;
        a.y = (((int)(p1 < pend)) & ((int)(s1 == h))) ? binv : 0.f;
        b.x = fcol[(size_t)n0 * N_DIM];
        b.y = fcol[(size_t)n1 * N_DIM];
        c = __builtin_amdgcn_wmma_f32_16x16x4_f32(false, a, false, b,
                                                  (short)0, c, false, false);
    }

    int mrow = (lane < 16) ? 0 : 8;
    int ncol = lane & 15;
    #pragma unroll
    for (int r = 0; r < 8; ++r) {
        int hh = h0 + r + mrow;
        if (hh < N_HEDGES)
            edge_raw[(size_t)hh * N_DIM + dimbase + ncol] = c[r];
    }
}

// ---------------------------------------------------------------- node phase: coalesced float4 gather
__global__ void hg_node_gather(const float* __restrict__ edge_raw,
                               const int* __restrict__ noff,
                               const int* __restrict__ nperm,
                               const int* __restrict__ edge_idx,
                               const float* __restrict__ Dsum,
                               const float* __restrict__ bias,
                               float* __restrict__ node_out) {
    int n = blockIdx.x;
    int d4 = blockIdx.y * blockDim.x + threadIdx.x;   // float4 index within row
    int b0 = noff[n], b1 = noff[n + 1];
    float4 acc = make_float4(0.f, 0.f, 0.f, 0.f);
    for (int p = b0; p < b1; ++p) {
        int e = edge_idx[nperm[p]];
        if (p + 1 < b1)
            __builtin_prefetch(&edge_raw[(size_t)edge_idx[nperm[p + 1]] * N_DIM + d4 * 4], 0, 0);
        float4 v = ((const float4*)(edge_raw + (size_t)e * N_DIM))[d4];
        acc.x += v.x; acc.y += v.y; acc.z += v.z; acc.w += v.w;
    }
    float ds = Dsum[n];
    float dinv = (ds == 0.f) ? 0.f : (1.f / ds);
    float4 bb = ((const float4*)bias)[d4];
    float4 o;
    o.x = acc.x * dinv + bb.x; o.x = (o.x > 0.f) ? o.x : NEG_SLOPE * o.x;
    o.y = acc.y * dinv + bb.y; o.y = (o.y > 0.f) ? o.y : NEG_SLOPE * o.y;
    o.z = acc.z * dinv + bb.z; o.z = (o.z > 0.f) ? o.z : NEG_SLOPE * o.z;
    o.w = acc.w * dinv + bb.w; o.w = (o.w > 0.f) ? o.w : NEG_SLOPE * o.w;
    ((float4*)(node_out + (size_t)n * N_DIM))[d4] = o;
}

// ---------------------------------------------------------------- in-place leaky on edge region (float4)
__global__ void hg_edge_leaky(float* edge_io) {
    int i = blockIdx.x * blockDim.x + threadIdx.x;
    if (i >= (N_HEDGES * N_DIM) / 4) return;
    float4 v = ((const float4*)edge_io)[i];
    v.x = (v.x > 0.f) ? v.x : NEG_SLOPE * v.x;
    v.y = (v.y > 0.f) ? v.y : NEG_SLOPE * v.y;
    v.z = (v.z > 0.f) ? v.z : NEG_SLOPE * v.z;
    v.w = (v.w > 0.f) ? v.w : NEG_SLOPE * v.w;
    ((float4*)edge_io)[i] = v;
}

// ----------------------------------------------------------------
extern "C" void kernel_launch(void* const* d_in, const int* in_sizes, int n_in,
                              void* d_out, int out_size, void* d_ws, size_t ws_size,
                              hipStream_t stream) {
    const float* feat     = (const float*)d_in[0];
    const int*   hidx     = (const int*)d_in[1];   // [2][5200]: row0 node, row1 edge
    const float* hw       = (const float*)d_in[3]; // hyperedge_weight (1000)
    const float* ew       = (const float*)d_in[6]; // EW_weight (5200)
    const float* bias     = (const float*)d_in[7]; // (4096)
    const int* node_idx = hidx;
    const int* edge_idx = hidx + N_INCID;

    // workspace carve-up (ints/floats, all 4B; ~140 KB total)
    char* ws = (char*)d_ws;
    int*   ecnt  = (int*)ws;                 ws += N_HEDGES * 4;
    int*   ncnt  = (int*)ws;                 ws += N_NODES * 4;
    int*   eoff  = (int*)ws;                 ws += (N_HEDGES + 1) * 4;
    int*   noff  = (int*)ws;                 ws += (N_NODES + 1) * 4;
    int*   ecur  = (int*)ws;                 ws += N_HEDGES * 4;
    int*   ncur  = (int*)ws;                 ws += N_NODES * 4;
    int*   eperm = (int*)ws;                 ws += N_INCID * 4;
    int*   nperm = (int*)ws;                 ws += N_INCID * 4;
    float* Dsum  = (float*)ws;               ws += N_NODES * 4;
    float* Bsum  = (float*)ws;               ws += N_HEDGES * 4;

    float* node_out = (float*)d_out;                           // 5200*4096
    float* edge_out = (float*)d_out + (size_t)N_NODES * N_DIM; // 1000*4096 (raw, then leaky)

    hg_init<<<(N_NODES + 255) / 256, 256, 0, stream>>>(Dsum, Bsum, ecnt, ncnt);
    hg_count<<<(N_INCID + 255) / 256, 256, 0, stream>>>(node_idx, edge_idx, hw, ew,
                                                        Dsum, Bsum, ecnt, ncnt);
    hg_scan<<<1, 256, 0, stream>>>(ecnt, eoff, ecur, N_HEDGES);
    hg_scan<<<1, 256, 0, stream>>>(ncnt, noff, ncur, N_NODES);
    hg_place<<<(N_INCID + 255) / 256, 256, 0, stream>>>(node_idx, edge_idx,
                                                        ecur, ncur, eperm, nperm);
    hg_sortseg<<<(N_HEDGES + N_NODES + 255) / 256, 256, 0, stream>>>(eperm, eoff, nperm, noff);

    dim3 egrid((N_HEDGES + 15) / 16, N_DIM / 64);          // 63 x 64, 4 waves/block
    hg_edge_wmma<<<egrid, 128, 0, stream>>>(feat, node_idx, edge_idx,
                                            eoff, eperm, Bsum, edge_out);

    dim3 ngrid(N_NODES, N_DIM / (256 * 4));                // float4 per thread
    hg_node_gather<<<ngrid, 256, 0, stream>>>(edge_out, noff, nperm, edge_idx,
                                              Dsum, bias, node_out);

    hg_edge_leaky<<<((N_HEDGES * N_DIM) / 4 + 255) / 256, 256, 0, stream>>>(edge_out);
}